// GatedLinearAttention_67095979099033
// MI455X (gfx1250) — compile-verified
//
#include <hip/hip_runtime.h>
#include <math.h>
#include <stdint.h>

// ---------------------------------------------------------------------------
// CDNA5 (gfx1250) fp32 WMMA: V_WMMA_F32_16X16X4_F32
// A: 16x4 f32 -> 2 VGPR/lane (v2f), B: 4x16 f32 -> v2f, C/D: 16x16 f32 -> v8f
// Per-lane layout (ISA 7.12.2):
//   a[j] = A[lane&15][k0 + 2*(lane>>4) + j]
//   b[j] = B[k0 + 2*(lane>>4) + j][lane&15]
//   c[i] = C[i + 8*(lane>>4)][lane&15]
// ---------------------------------------------------------------------------
typedef float v8f __attribute__((ext_vector_type(8)));
typedef float v2f __attribute__((ext_vector_type(2)));

__device__ __forceinline__ v8f wmma_f32(v2f a, v2f b, v8f c) {
  return __builtin_amdgcn_wmma_f32_16x16x4_f32(false, a, false, b,
                                               (short)0, c, false, false);
}

__device__ __forceinline__ v8f zero8() {
  v8f z = {0.f, 0.f, 0.f, 0.f, 0.f, 0.f, 0.f, 0.f};
  return z;
}

// ---------------------------------------------------------------------------
// Async global->LDS copy (GLOBAL_LOAD_ASYNC_TO_LDS_B128, ASYNCcnt-tracked).
// Builtin signature (from clang diagnostic): param 1 is
// 'int __vector(4) __device__ *' i.e. v4i in address space 1 (global);
// param 2 is the LDS destination (address space 3), then offset, cpol.
// ---------------------------------------------------------------------------
#if __has_builtin(__builtin_amdgcn_global_load_async_to_lds_b128)
#define GLA_HAVE_ASYNC_LDS 1
typedef int v4i __attribute__((vector_size(16)));
typedef __attribute__((address_space(1))) v4i* gbl_v4i_p;
typedef __attribute__((address_space(3))) v4i* lds_v4i_p;

__device__ __forceinline__ void async_copy_b128(const void* g, void* l) {
  // Low 32 bits of a flat shared address are the LDS offset (aperture map).
  __builtin_amdgcn_global_load_async_to_lds_b128(
      (gbl_v4i_p)(uintptr_t)g, (lds_v4i_p)(uintptr_t)l, 0, 0);
}
#else
#define GLA_HAVE_ASYNC_LDS 0
#endif

__device__ __forceinline__ void wait_async_lds() {
#if GLA_HAVE_ASYNC_LDS
#if __has_builtin(__builtin_amdgcn_s_wait_asynccnt)
  __builtin_amdgcn_s_wait_asynccnt(0);
#else
  asm volatile("s_wait_asynccnt 0" ::: "memory");
#endif
#endif
}

// ---------------------------------------------------------------------------
// Generic C = act(A[MxK] @ W[KxN] + bias) GEMM.
// Block tile 64x64, 256 threads (8 waves), each wave owns two 16x16 subtiles.
// K staged 16 at a time via async global->LDS B128 copies.
//   As[64][20] : A tile, row-major (stride 20 -> 80B rows keep 16B alignment)
//   Ws[16][68] : W tile, row-major (stride 68 -> 272B rows, 16B aligned)
// A frag = contiguous float2; B frag = two b32 reads one LDS row apart
// (merged by the compiler into ds_load_2addr_b32).
// ACT: 0 = none, 1 = relu, 2 = log_sigmoid(x)/16
// ---------------------------------------------------------------------------
template <int ACT>
__global__ __launch_bounds__(256)
void gemm_bias_act(const float* __restrict__ A, const float* __restrict__ W,
                   const float* __restrict__ bias, float* __restrict__ C,
                   int M, int N, int K) {
  __shared__ float As[64][20];
  __shared__ float Ws[16][68];

  const int tid  = threadIdx.x;
  const int lane = tid & 31;
  const int wave = tid >> 5;
  const int lo   = lane & 15;
  const int hi   = lane >> 4;
  const int m0   = blockIdx.x * 64;
  const int n0   = blockIdx.y * 64;
  const int sy   = wave >> 2;          // 0..1
  const int sx   = wave & 3;           // 0..3
  const int coln = sx * 16 + lo;       // column within block tile

  v8f acc0 = zero8();
  v8f acc1 = zero8();

  const int i  = tid * 4;              // element index 0..1023, 4 per thread
  const int r  = i >> 4, cc = i & 15;  // A-tile coords (64x16)
  const int kr = i >> 6, nc = i & 63;  // W-tile coords (16x64)

  for (int k0 = 0; k0 < K; k0 += 16) {
#if GLA_HAVE_ASYNC_LDS
    async_copy_b128(&A[(size_t)(m0 + r) * K + (k0 + cc)], &As[r][cc]);
    if (n0 + nc < N)  // N%4==0, nc%4==0 -> whole 16B in bounds; OOB lanes off
      async_copy_b128(&W[(size_t)(k0 + kr) * N + (n0 + nc)], &Ws[kr][nc]);
    wait_async_lds();
#else
    {
      const float4 av = *(const float4*)&A[(size_t)(m0 + r) * K + (k0 + cc)];
      *(float4*)&As[r][cc] = av;
      float4 wv = {0.f, 0.f, 0.f, 0.f};
      if (n0 + nc < N)
        wv = *(const float4*)&W[(size_t)(k0 + kr) * N + (n0 + nc)];
      *(float4*)&Ws[kr][nc] = wv;
    }
#endif
    __syncthreads();
#pragma unroll
    for (int kk = 0; kk < 16; kk += 4) {
      v2f b0;
      b0[0] = Ws[kk + 2 * hi + 0][coln];
      b0[1] = Ws[kk + 2 * hi + 1][coln];
      const v2f a0 = *(const v2f*)&As[sy * 16 + lo][kk + 2 * hi];
      acc0 = wmma_f32(a0, b0, acc0);
      const v2f a1 = *(const v2f*)&As[(sy + 2) * 16 + lo][kk + 2 * hi];
      acc1 = wmma_f32(a1, b0, acc1);
    }
    __syncthreads();
  }

  const int col = n0 + coln;
  if (col < N) {
    const float bv = bias ? bias[col] : 0.f;
#pragma unroll
    for (int half = 0; half < 2; ++half) {
      const v8f acc = half ? acc1 : acc0;
      const int rbase = m0 + (sy + 2 * half) * 16 + 8 * hi;
#pragma unroll
      for (int j = 0; j < 8; ++j) {
        float x = acc[j] + bv;
        if (ACT == 1) x = fmaxf(x, 0.f);
        if (ACT == 2)
          x = (fminf(x, 0.f) - log1pf(__expf(-fabsf(x)))) * (1.0f / 16.0f);
        C[(size_t)(rbase + j) * N + col] = x;
      }
    }
  }
}

// ---------------------------------------------------------------------------
// Chunkwise GLA recurrence. One workgroup per (b,h); GQA n_rep=2.
// Dynamic LDS (~244 KB, fits 320 KB WGP LDS):
//   St[128][130]  : state, TRANSPOSED St[e][d]  (B-operand of q_@S)
//   qe[64][128]   : q * exp(b) * D^-0.5         (A-operand)
//   ke[64][128]   : k * exp(-b)                 (B-operand of q_@k^T, rows=t')
//   kendt[128][66]: (k * exp(b_end-b))^T        (A-operand of state update)
//   vt[128][66]   : v^T                         (B-operand)
//   bs[64][128]   : inclusive cumsum of gates
//   As[64][66]    : causal-masked scores        (A-operand of A@v)
// ---------------------------------------------------------------------------
#define GLA_SMEM_FLOATS (128 * 130 + 64 * 128 + 64 * 128 + 128 * 66 + \
                         128 * 66 + 64 * 128 + 64 * 66)

__global__ __launch_bounds__(256)
void gla_chunk_kernel(const float* __restrict__ q, const float* __restrict__ k,
                      const float* __restrict__ v, const float* __restrict__ g,
                      float* __restrict__ o) {
  extern __shared__ float sm[];
  float* St    = sm;                   // 128*130
  float* qe    = St + 128 * 130;       // 64*128
  float* ke    = qe + 64 * 128;        // 64*128
  float* kendt = ke + 64 * 128;        // 128*66
  float* vt    = kendt + 128 * 66;     // 128*66
  float* bs    = vt + 128 * 66;        // 64*128
  float* As    = bs + 64 * 128;        // 64*66

  const int tid  = threadIdx.x;
  const int lane = tid & 31;
  const int wave = tid >> 5;
  const int lo   = lane & 15;
  const int hi   = lane >> 4;
  const int bh = blockIdx.x;
  const int b  = bh >> 4;
  const int h  = bh & 15;
  const int hk = h >> 1;               // 16 q-heads -> 8 kv-heads
  const float qscale = 0.08838834764831845f;  // 128^-0.5

  for (int i = tid; i < 128 * 130; i += 256) St[i] = 0.f;
  __syncthreads();

  for (int c = 0; c < 32; ++c) {
    const size_t t0 = (size_t)b * 2048 + (size_t)c * 64;  // flat token row

    // ---- step 0: inclusive cumsum of log-gates along time ----
    if (tid < 128) {
      float accum = 0.f;
      for (int t = 0; t < 64; ++t) {
        accum += g[(t0 + t) * 1024 + hk * 128 + tid];
        bs[t * 128 + tid] = accum;
      }
    }
    __syncthreads();

    // ---- step 1: stage decorated operands ----
    for (int idx = tid; idx < 64 * 128; idx += 256) {
      const int t = idx >> 7, d = idx & 127;
      const float bval = bs[idx];
      const float bend = bs[63 * 128 + d];
      const float qv = q[(t0 + t) * 2048 + h * 128 + d];
      const float kv = k[(t0 + t) * 1024 + hk * 128 + d];
      const float vv = v[(t0 + t) * 1024 + hk * 128 + d];
      qe[idx] = qv * __expf(bval) * qscale;
      ke[idx] = kv * __expf(-bval);
      kendt[d * 66 + t] = kv * __expf(bend - bval);
      vt[d * 66 + t] = vv;
    }
    __syncthreads();

    // ---- step 2: A = qe @ ke^T (K=128), causal mask, store to As ----
    for (int s = 0; s < 2; ++s) {
      const int tt = wave + s * 8;     // 16 tiles of 16x16 over [64x64]
      const int ti = tt >> 2, tj = tt & 3;
      v8f acc = zero8();
#pragma unroll 4
      for (int kk = 0; kk < 128; kk += 4) {
        const v2f a  = *(const v2f*)&qe[(ti * 16 + lo) * 128 + kk + 2 * hi];
        const v2f bb = *(const v2f*)&ke[(tj * 16 + lo) * 128 + kk + 2 * hi];
        acc = wmma_f32(a, bb, acc);
      }
#pragma unroll
      for (int j = 0; j < 8; ++j) {
        const int row = ti * 16 + j + 8 * hi;
        const int col = tj * 16 + lo;
        As[row * 66 + col] = (col <= row) ? acc[j] : 0.f;
      }
    }
    __syncthreads();

    // ---- step 3: o = As @ v + qe @ S  (reads old S) ----
    {
      const int ti = wave >> 1;        // 4 t-row blocks x 8 e-col blocks
      for (int e4 = 0; e4 < 4; ++e4) {
        const int te = (wave & 1) * 4 + e4;
        v8f acc = zero8();
#pragma unroll 4
        for (int kk = 0; kk < 64; kk += 4) {      // K = 64 (t')
          const v2f a  = *(const v2f*)&As[(ti * 16 + lo) * 66 + kk + 2 * hi];
          const v2f bb = *(const v2f*)&vt[(te * 16 + lo) * 66 + kk + 2 * hi];
          acc = wmma_f32(a, bb, acc);
        }
#pragma unroll 4
        for (int kk = 0; kk < 128; kk += 4) {     // K = 128 (d)
          const v2f a  = *(const v2f*)&qe[(ti * 16 + lo) * 128 + kk + 2 * hi];
          const v2f bb = *(const v2f*)&St[(te * 16 + lo) * 130 + kk + 2 * hi];
          acc = wmma_f32(a, bb, acc);
        }
#pragma unroll
        for (int j = 0; j < 8; ++j) {
          const int t = ti * 16 + j + 8 * hi;
          const int e = te * 16 + lo;
          o[(t0 + t) * 2048 + h * 128 + e] = acc[j];
        }
      }
    }
    __syncthreads();

    // ---- step 4: S = diag(exp(b_end)) S + kend^T @ v ----
    {
      const int td = wave;             // 8 d-blocks x 8 e-blocks
      for (int te = 0; te < 8; ++te) {
        v8f acc = zero8();
#pragma unroll 4
        for (int kk = 0; kk < 64; kk += 4) {      // K = 64 (t)
          const v2f a  = *(const v2f*)&kendt[(td * 16 + lo) * 66 + kk + 2 * hi];
          const v2f bb = *(const v2f*)&vt[(te * 16 + lo) * 66 + kk + 2 * hi];
          acc = wmma_f32(a, bb, acc);
        }
#pragma unroll
        for (int j = 0; j < 8; ++j) {
          const int d = td * 16 + j + 8 * hi;
          const int e = te * 16 + lo;
          const float sc = __expf(bs[63 * 128 + d]);
          St[e * 130 + d] = St[e * 130 + d] * sc + acc[j];
        }
      }
    }
    __syncthreads();
  }
}

// ---------------------------------------------------------------------------
// Per-head RMSNorm over D=128, in place. One wave per 128-element segment.
// ---------------------------------------------------------------------------
__global__ __launch_bounds__(256)
void rmsnorm_kernel(float* __restrict__ o, const float* __restrict__ w) {
  const int lane = threadIdx.x & 31;
  const int wave = threadIdx.x >> 5;
  const size_t seg = (size_t)blockIdx.x * 8 + wave;
  float* p = o + seg * 128 + lane * 4;
  float4 x = *(float4*)p;
  float ss = x.x * x.x + x.y * x.y + x.z * x.z + x.w * x.w;
#pragma unroll
  for (int off = 16; off > 0; off >>= 1) ss += __shfl_xor(ss, off, 32);
  const float r = rsqrtf(ss * (1.0f / 128.0f) + 1e-5f);
  const float4 wv = *(const float4*)&w[lane * 4];
  x.x *= r * wv.x; x.y *= r * wv.y; x.z *= r * wv.z; x.w *= r * wv.w;
  *(float4*)p = x;
}

// ---------------------------------------------------------------------------
// Launcher. Inputs (all f32): hs, Wq, bq, Wk, bk, Wv, bv, Wg1, Wg2, bg2,
// g_norm_w, Wo. Output: [4096, 2048] f32.
// ---------------------------------------------------------------------------
extern "C" void kernel_launch(void* const* d_in, const int* in_sizes, int n_in,
                              void* d_out, int out_size, void* d_ws,
                              size_t ws_size, hipStream_t stream) {
  (void)in_sizes; (void)n_in; (void)out_size; (void)ws_size;
  const float* hs  = (const float*)d_in[0];
  const float* Wq  = (const float*)d_in[1];
  const float* bq  = (const float*)d_in[2];
  const float* Wk  = (const float*)d_in[3];
  const float* bk  = (const float*)d_in[4];
  const float* Wv  = (const float*)d_in[5];
  const float* bv  = (const float*)d_in[6];
  const float* Wg1 = (const float*)d_in[7];
  const float* Wg2 = (const float*)d_in[8];
  const float* bg2 = (const float*)d_in[9];
  const float* gnw = (const float*)d_in[10];
  const float* Wo  = (const float*)d_in[11];
  float* out = (float*)d_out;

  float* ws = (float*)d_ws;
  float* q  = ws;                        // 4096*2048
  float* k  = q  + (size_t)4096 * 2048;  // 4096*1024
  float* v  = k  + (size_t)4096 * 1024;  // 4096*1024
  float* g  = v  + (size_t)4096 * 1024;  // 4096*1024
  float* g1 = g  + (size_t)4096 * 1024;  // 4096*16
  float* o  = g1 + (size_t)4096 * 16;    // 4096*2048

  const dim3 blk(256);
  // Projections (f32 WMMA GEMMs, async LDS staging)
  gemm_bias_act<1><<<dim3(64, 32), blk, 0, stream>>>(hs, Wq, bq, q, 4096, 2048, 2048);
  gemm_bias_act<1><<<dim3(64, 16), blk, 0, stream>>>(hs, Wk, bk, k, 4096, 1024, 2048);
  gemm_bias_act<0><<<dim3(64, 16), blk, 0, stream>>>(hs, Wv, bv, v, 4096, 1024, 2048);
  // Low-rank gate: g1 = hs @ Wg1 ; g = log_sigmoid(g1 @ Wg2 + bg2)/16
  gemm_bias_act<0><<<dim3(64, 1),  blk, 0, stream>>>(hs, Wg1, nullptr, g1, 4096, 16, 2048);
  gemm_bias_act<2><<<dim3(64, 16), blk, 0, stream>>>(g1, Wg2, bg2, g, 4096, 1024, 16);
  // Chunked GLA recurrence: one workgroup per (b,h)
  gla_chunk_kernel<<<32, blk, GLA_SMEM_FLOATS * sizeof(float), stream>>>(q, k, v, g, o);
  // Per-head RMSNorm (4096*16 segments of 128)
  rmsnorm_kernel<<<8192, blk, 0, stream>>>(o, gnw);
  // Output projection
  gemm_bias_act<0><<<dim3(64, 32), blk, 0, stream>>>(o, Wo, nullptr, out, 4096, 2048, 2048);
}